// SingleQubitLayer_52183852646869
// MI455X (gfx1250) — compile-verified
//
#include <hip/hip_runtime.h>

// CDNA5 / gfx1250, wave32.
// z = k0 + k1*cos(x0) + k2*sin(x0), evaluated per 16-sample tile with
// V_WMMA_F32_16X16X4_F32:  A(16x4) = [1, cos, sin, 0] rows,
// B(4x16) columns all = [k0,k1,k2,0]^T  =>  D[m][n] = z_m for all n.

typedef __attribute__((ext_vector_type(2))) float v2f;
typedef __attribute__((ext_vector_type(8))) float v8f;

struct cplx { float re, im; };
__device__ inline cplx cmul(cplx a, cplx b) {
    return { a.re * b.re - a.im * b.im, a.re * b.im + a.im * b.re };
}
__device__ inline cplx cadd(cplx a, cplx b) { return { a.re + b.re, a.im + b.im }; }
__device__ inline float mag2(cplx a) { return a.re * a.re + a.im * a.im; }

__device__ inline void mm2(const cplx A[2][2], const cplx B[2][2], cplx C[2][2]) {
    for (int i = 0; i < 2; ++i)
        for (int j = 0; j < 2; ++j)
            C[i][j] = cadd(cmul(A[i][0], B[0][j]), cmul(A[i][1], B[1][j]));
}

// Fold the batch-shared 2x2 unitary into (k0,k1,k2). One thread; runs in ~ns.
__global__ void sq_coeff_kernel(const float* __restrict__ w, int n_layers,
                                float* __restrict__ k) {
    if (threadIdx.x != 0 || blockIdx.x != 0) return;
    cplx M[2][2] = { { {1.f, 0.f}, {0.f, 0.f} }, { {0.f, 0.f}, {1.f, 0.f} } };
    for (int l = 0; l < n_layers; ++l) {
        float tx = w[3 * l + 0] * 0.5f;
        float ty = w[3 * l + 1] * 0.5f;
        float tz = w[3 * l + 2] * 0.5f;
        float cx = cosf(tx), sx = sinf(tx);
        float cy = cosf(ty), sy = sinf(ty);
        float cz = cosf(tz), sz = sinf(tz);
        // RX = [[c, -i s], [-i s, c]];  RY = [[c, -s], [s, c]];
        // RZ = diag(e^{-i t/2}, e^{+i t/2})
        cplx RX[2][2] = { { {cx, 0.f}, {0.f, -sx} }, { {0.f, -sx}, {cx, 0.f} } };
        cplx RY[2][2] = { { {cy, 0.f}, {-sy, 0.f} }, { {sy, 0.f}, {cy, 0.f} } };
        cplx RZ[2][2] = { { {cz, -sz}, {0.f, 0.f} }, { {0.f, 0.f}, {cz, sz} } };
        cplx T[2][2], G[2][2], Mn[2][2];
        mm2(RY, RX, T);
        mm2(RZ, T, G);
        mm2(G, M, Mn);
        for (int i = 0; i < 2; ++i)
            for (int j = 0; j < 2; ++j) M[i][j] = Mn[i][j];
    }
    float P = mag2(M[0][0]) - mag2(M[1][0]);
    float Q = mag2(M[0][1]) - mag2(M[1][1]);
    float R = 2.f * ((M[0][0].re * M[0][1].re + M[0][0].im * M[0][1].im)
                   - (M[1][0].re * M[1][1].re + M[1][0].im * M[1][1].im));
    k[0] = 0.5f * (P + Q);
    k[1] = 0.5f * (P - Q);
    k[2] = 0.5f * R;
}

// One wave32 handles one 16-sample tile via a single f32 WMMA.
// A 16x4 layout (ISA 7.12.2): lanes 0-15 hold K=0 (v0) and K=1 (v1) for M=lane;
// lanes 16-31 hold K=2 (v0) and K=3 (v1). B 4x16 mirrors the K split across
// half-waves (row K striped across lanes within a VGPR).
// D 16x16 f32: VGPR r, lanes 0-15 -> row r, lanes 16-31 -> row 8+r; since every
// column of D equals z, lane 0's 8 acc VGPRs = z_0..z_7, lane 16's = z_8..z_15.
__global__ __launch_bounds__(256) void sq_wmma_kernel(const float* __restrict__ x,
                                                      const float* __restrict__ k,
                                                      float* __restrict__ out,
                                                      int B) {
    const int lane = threadIdx.x & 31;
    const int wave = (int)((blockIdx.x * blockDim.x + threadIdx.x) >> 5);
    const int base = wave * 16;
    if (base >= B) return;  // uniform per wave

    const float k0 = k[0];
    const float k1 = k[1];
    const float k2 = k[2];

    int idx = base + (lane & 15);
    if (idx >= B) idx = B - 1;                 // tail clamp (load only)
    const float x0 = x[(size_t)idx * 30u];     // stride-30 column-0 read
    float s, c;
    sincosf(x0, &s, &c);

    const bool lo = (lane < 16);
    v2f a, b;
    a.x = lo ? 1.0f : s;    // A: K=0 -> 1,      K=2 -> sin
    a.y = lo ? c    : 0.0f; //    K=1 -> cos,    K=3 -> 0
    b.x = lo ? k0   : k2;   // B: K=0 -> k0,     K=2 -> k2
    b.y = lo ? k1   : 0.0f; //    K=1 -> k1,     K=3 -> 0

    v8f acc = {};
    acc = __builtin_amdgcn_wmma_f32_16x16x4_f32(
        /*neg_a=*/false, a, /*neg_b=*/false, b,
        /*c_mod=*/(short)0, acc, /*reuse_a=*/false, /*reuse_b=*/false);

    if ((lane & 15) == 0) {                    // lanes 0 and 16 write the tile
        const int o = base + (lane >> 1);      // +0 or +8
        if (o + 8 <= B) {
            *(v8f*)(out + o) = acc;            // 32B store (64B/32B aligned)
        } else {
#pragma unroll
            for (int j = 0; j < 8; ++j)
                if (o + j < B) out[o + j] = acc[j];
        }
    }
}

extern "C" void kernel_launch(void* const* d_in, const int* in_sizes, int n_in,
                              void* d_out, int out_size, void* d_ws, size_t ws_size,
                              hipStream_t stream) {
    const float* x = (const float*)d_in[0];   // [B, 30] float32
    const float* w = (const float*)d_in[1];   // [n_layers, 3] float32
    float* out = (float*)d_out;               // [B, 1] float32
    float* ws  = (float*)d_ws;                // 3 floats of scratch

    const int B = in_sizes[0] / 30;
    const int n_layers = in_sizes[1] / 3;

    sq_coeff_kernel<<<1, 32, 0, stream>>>(w, n_layers, ws);

    const int tiles = (B + 15) / 16;          // one wave per 16-sample tile
    const int wavesPerBlock = 256 / 32;       // 8 waves per block
    const int blocks = (tiles + wavesPerBlock - 1) / wavesPerBlock;
    sq_wmma_kernel<<<blocks, 256, 0, stream>>>(x, ws, out, B);
}